// Meshes_36146444763461
// MI455X (gfx1250) — compile-verified
//
#include <hip/hip_runtime.h>
#include <hip/hip_bf16.h>

// ---------------------------------------------------------------------------
// Mesh pipeline for MI455X (gfx1250).
// verts: (V,3) f32, faces: (F,3) i32.  Outputs (concatenated float):
//   vn (3V) | edges (3F*2, int values as float) | lap (6F) | cot (3F) | inv_areas (V)
//
// Design: L2-resident gathers/atomics (verts=24MB, deg/counts=8MB each, all
// << 192MB L2), counting sort over h = min*V + max buckets instead of a
// 12M-key radix sort, and CDNA5 TDM (tensor_load_to_lds + s_wait_tensorcnt)
// for the streaming faces tile into LDS.
// ---------------------------------------------------------------------------

#define FACE_TILE   256
#define SCAN_BLOCK  256
#define SCAN_ITEMS  16
#define SCAN_TILE   (SCAN_BLOCK * SCAN_ITEMS)   // 4096

typedef unsigned int  tdm_u32x4 __attribute__((ext_vector_type(4)));
typedef int           tdm_i32x8 __attribute__((ext_vector_type(8)));
typedef int           tdm_i32x4 __attribute__((ext_vector_type(4)));

// ---------------------------------------------------------------------------
// Kernel 1: per-face pass.
//  - TDM async DMA of the block's face-index tile into LDS (wave 0 issues,
//    waits on TENSORcnt, then workgroup barrier publishes the tile).
//  - cross-product face normal -> float atomics into vn
//  - Heron area -> cot row + float atomics into area_sum
//  - 3 sorted edge pairs -> bucket histogram counts[min] (u32 atomics, L2)
// ---------------------------------------------------------------------------
__global__ void face_kernel(const float* __restrict__ verts,
                            const int*   __restrict__ faces,
                            float*       __restrict__ vn,        // 3V, pre-zeroed
                            float*       __restrict__ cot,       // 3F
                            float*       __restrict__ area_sum,  // V, pre-zeroed
                            unsigned*    __restrict__ counts,    // V, pre-zeroed
                            int V, int F) {
    __shared__ int sf[3 * FACE_TILE];

    const int blockStart = blockIdx.x * FACE_TILE;
    const int nface = min(FACE_TILE, F - blockStart);
    const int nelem = nface * 3;

#if defined(__gfx1250__) && __has_builtin(__builtin_amdgcn_tensor_load_to_lds) && __has_builtin(__builtin_amdgcn_s_wait_tensorcnt)
    if (threadIdx.x < 32) {  // wave 0 issues the DMA (EXEC ignored by TDM)
        unsigned long long gaddr =
            (unsigned long long)(const void*)(faces + (size_t)blockStart * 3);
        unsigned lds_addr = (unsigned)(unsigned long long)(void*)sf; // low 32 bits = LDS offset
        unsigned remain = (unsigned)(3 * (F - blockStart));          // elements left in tensor

        tdm_u32x4 g0;
        g0[0] = 1u;                                    // count=1, user mode, no gather
        g0[1] = lds_addr;                              // lds_addr (bytes)
        g0[2] = (unsigned)gaddr;                       // global_addr[31:0]
        g0[3] = (unsigned)((gaddr >> 32) & 0x1FFFFFFull) | (2u << 30); // addr[56:32] | type=2

        tdm_i32x8 g1;
        g1[0] = (int)(2u << 16);                       // data_size = 2 (4 bytes)
        g1[1] = (int)((remain & 0xFFFFu) << 16);       // tensor_dim0[15:0]
        g1[2] = (int)(((remain >> 16) & 0xFFFFu)       // tensor_dim0[31:16]
                      | (1u << 16));                   // tensor_dim1 = 1
        g1[3] = (int)(((unsigned)nelem) << 16);        // tile_dim0 = nelem (1-D tile)
        g1[4] = 0;                                     // tile_dim1 = tile_dim2 = 0
        g1[5] = (int)remain;                           // tensor_dim0_stride[31:0]
        g1[6] = 0;                                     // stride hi / dim1_stride lo
        g1[7] = 0;

        tdm_i32x4 g2 = {0, 0, 0, 0};
        tdm_i32x4 g3 = {0, 0, 0, 0};
        tdm_i32x8 g4 = {0, 0, 0, 0, 0, 0, 0, 0};

        __builtin_amdgcn_tensor_load_to_lds(g0, g1, g2, g3, g4, 0);
        __builtin_amdgcn_s_wait_tensorcnt(0);
    }
#else
    for (int i = threadIdx.x; i < nelem; i += blockDim.x)
        sf[i] = faces[(size_t)blockStart * 3 + i];
#endif
    __syncthreads();

    if ((int)threadIdx.x >= nface) return;
    const int f = blockStart + threadIdx.x;

    const int a = sf[threadIdx.x * 3 + 0];
    const int b = sf[threadIdx.x * 3 + 1];
    const int c = sf[threadIdx.x * 3 + 2];

    const float* p0 = verts + 3 * (size_t)a;   // v0
    const float* p1 = verts + 3 * (size_t)b;   // v1
    const float* p2 = verts + 3 * (size_t)c;   // v2
    const float v0x = p0[0], v0y = p0[1], v0z = p0[2];
    const float v1x = p1[0], v1y = p1[1], v1z = p1[2];
    const float v2x = p2[0], v2y = p2[1], v2z = p2[2];

    // face_n = cross(v2 - v1, v0 - v1)
    const float e1x = v2x - v1x, e1y = v2y - v1y, e1z = v2z - v1z;
    const float e2x = v0x - v1x, e2y = v0y - v1y, e2z = v0z - v1z;
    const float nx = e1y * e2z - e1z * e2y;
    const float ny = e1z * e2x - e1x * e2z;
    const float nz = e1x * e2y - e1y * e2x;

    atomicAdd(&vn[3 * (size_t)a + 0], nx);
    atomicAdd(&vn[3 * (size_t)a + 1], ny);
    atomicAdd(&vn[3 * (size_t)a + 2], nz);
    atomicAdd(&vn[3 * (size_t)b + 0], nx);
    atomicAdd(&vn[3 * (size_t)b + 1], ny);
    atomicAdd(&vn[3 * (size_t)b + 2], nz);
    atomicAdd(&vn[3 * (size_t)c + 0], nx);
    atomicAdd(&vn[3 * (size_t)c + 1], ny);
    atomicAdd(&vn[3 * (size_t)c + 2], nz);

    // edge lengths: A=|v1-v2|, B=|v0-v2|, C=|v0-v1|
    const float dAx = v1x - v2x, dAy = v1y - v2y, dAz = v1z - v2z;
    const float dBx = v0x - v2x, dBy = v0y - v2y, dBz = v0z - v2z;
    const float dCx = v0x - v1x, dCy = v0y - v1y, dCz = v0z - v1z;
    const float A2 = dAx * dAx + dAy * dAy + dAz * dAz;
    const float B2 = dBx * dBx + dBy * dBy + dBz * dBz;
    const float C2 = dCx * dCx + dCy * dCy + dCz * dCz;
    const float A = sqrtf(A2), B = sqrtf(B2), C = sqrtf(C2);
    const float s = 0.5f * (A + B + C);
    float t = s * (s - A) * (s - B) * (s - C);
    t = fmaxf(t, 1e-12f);
    const float area = sqrtf(t);
    const float inv4a = 0.25f / area;

    cot[3 * (size_t)f + 0] = (B2 + C2 - A2) * inv4a;
    cot[3 * (size_t)f + 1] = (A2 + C2 - B2) * inv4a;
    cot[3 * (size_t)f + 2] = (A2 + B2 - C2) * inv4a;

    atomicAdd(&area_sum[a], area);
    atomicAdd(&area_sum[b], area);
    atomicAdd(&area_sum[c], area);

    // edge bucket histogram: pairs (b,c), (c,a), (a,b) keyed by min vertex
    {
        int mn = min(b, c); atomicAdd(&counts[mn], 1u);
        mn = min(c, a);     atomicAdd(&counts[mn], 1u);
        mn = min(a, b);     atomicAdd(&counts[mn], 1u);
    }
}

// ---------------------------------------------------------------------------
// Hierarchical exclusive scan (u32): tile scan -> single-block partials scan
// (optionally reporting the grand total) -> per-element add.
// ---------------------------------------------------------------------------
__global__ void scan_tile_kernel(const unsigned* __restrict__ in,
                                 unsigned* __restrict__ out,
                                 unsigned* __restrict__ partials, int n) {
    __shared__ unsigned sh[SCAN_BLOCK];
    const int base = blockIdx.x * SCAN_TILE + threadIdx.x * SCAN_ITEMS;
    unsigned v[SCAN_ITEMS];
    unsigned sum = 0;
    for (int k = 0; k < SCAN_ITEMS; ++k) {
        const int i = base + k;
        const unsigned x = (i < n) ? in[i] : 0u;
        v[k] = sum;
        sum += x;
    }
    sh[threadIdx.x] = sum;
    __syncthreads();
    for (int off = 1; off < SCAN_BLOCK; off <<= 1) {
        const unsigned tval = (threadIdx.x >= (unsigned)off) ? sh[threadIdx.x - off] : 0u;
        __syncthreads();
        sh[threadIdx.x] += tval;
        __syncthreads();
    }
    const unsigned texcl = sh[threadIdx.x] - sum;
    for (int k = 0; k < SCAN_ITEMS; ++k) {
        const int i = base + k;
        if (i < n) out[i] = texcl + v[k];
    }
    if (threadIdx.x == SCAN_BLOCK - 1) partials[blockIdx.x] = sh[SCAN_BLOCK - 1];
}

__global__ void scan_small_kernel(unsigned* __restrict__ data, int m,
                                  unsigned* __restrict__ total_out) {
    __shared__ unsigned sh[SCAN_BLOCK];
    const int t = threadIdx.x;
    const int base = t * SCAN_ITEMS;
    unsigned v[SCAN_ITEMS];
    unsigned sum = 0;
    for (int k = 0; k < SCAN_ITEMS; ++k) {
        const int i = base + k;
        const unsigned x = (i < m) ? data[i] : 0u;
        v[k] = sum;
        sum += x;
    }
    sh[t] = sum;
    __syncthreads();
    for (int off = 1; off < SCAN_BLOCK; off <<= 1) {
        const unsigned tval = (t >= off) ? sh[t - off] : 0u;
        __syncthreads();
        sh[t] += tval;
        __syncthreads();
    }
    const unsigned texcl = sh[t] - sum;
    for (int k = 0; k < SCAN_ITEMS; ++k) {
        const int i = base + k;
        if (i < m) data[i] = texcl + v[k];
    }
    if (t == SCAN_BLOCK - 1 && total_out) *total_out = sh[SCAN_BLOCK - 1];
}

__global__ void scan_add_kernel(unsigned* __restrict__ out,
                                const unsigned* __restrict__ partials, int n) {
    const int i = blockIdx.x * blockDim.x + threadIdx.x;
    if (i < n) out[i] += partials[i / SCAN_TILE];
}

// ---------------------------------------------------------------------------
// Kernel 3: scatter each directed edge's max vertex into its min-bucket.
// ---------------------------------------------------------------------------
__global__ void scatter_kernel(const int* __restrict__ faces,
                               const unsigned* __restrict__ offsets,
                               unsigned* __restrict__ cursor,   // pre-zeroed
                               unsigned* __restrict__ smax, int F) {
    const int f = blockIdx.x * blockDim.x + threadIdx.x;
    if (f >= F) return;
    const int a = faces[3 * (size_t)f + 0];
    const int b = faces[3 * (size_t)f + 1];
    const int c = faces[3 * (size_t)f + 2];
    int mn, mx;
    mn = min(b, c); mx = max(b, c);
    smax[offsets[mn] + atomicAdd(&cursor[mn], 1u)] = (unsigned)mx;
    mn = min(c, a); mx = max(c, a);
    smax[offsets[mn] + atomicAdd(&cursor[mn], 1u)] = (unsigned)mx;
    mn = min(a, b); mx = max(a, b);
    smax[offsets[mn] + atomicAdd(&cursor[mn], 1u)] = (unsigned)mx;
}

// ---------------------------------------------------------------------------
// Kernel 4: per-bucket insertion sort + in-place dedup (buckets avg ~6 long,
// L2-resident).  Produces ascending (min, max) == ascending hash order.
// ---------------------------------------------------------------------------
__global__ void bucket_kernel(const unsigned* __restrict__ offsets,
                              const unsigned* __restrict__ counts,
                              unsigned* __restrict__ smax,
                              unsigned* __restrict__ ucount, int V) {
    const int v = blockIdx.x * blockDim.x + threadIdx.x;
    if (v >= V) return;
    const unsigned base = offsets[v];
    const unsigned len  = counts[v];
    for (unsigned i = 1; i < len; ++i) {
        const unsigned key = smax[base + i];
        int j = (int)i - 1;
        while (j >= 0 && smax[base + j] > key) {
            smax[base + j + 1] = smax[base + j];
            --j;
        }
        smax[base + j + 1] = key;
    }
    unsigned u = 0;
    for (unsigned i = 0; i < len; ++i) {
        const unsigned x = smax[base + i];
        if (u == 0 || smax[base + u - 1] != x) {
            smax[base + u] = x;
            ++u;
        }
    }
    ucount[v] = u;
}

// ---------------------------------------------------------------------------
// Kernel 5: emit unique edges (as float-valued ints) + degree accumulation.
// ---------------------------------------------------------------------------
__global__ void emit_kernel(const unsigned* __restrict__ offsets,
                            const unsigned* __restrict__ ucount,
                            const unsigned* __restrict__ uoffsets,
                            const unsigned* __restrict__ smax,
                            float* __restrict__ edges_out,   // 2*3F, pre-zeroed
                            float* __restrict__ deg,         // V, pre-zeroed
                            int V) {
    const int v = blockIdx.x * blockDim.x + threadIdx.x;
    if (v >= V) return;
    const unsigned base = offsets[v];
    const unsigned u    = ucount[v];
    const unsigned uo   = uoffsets[v];
    for (unsigned j = 0; j < u; ++j) {
        const unsigned mx  = smax[base + j];
        const unsigned pos = uo + j;
        edges_out[2 * (size_t)pos + 0] = (float)v;
        edges_out[2 * (size_t)pos + 1] = (float)mx;
        atomicAdd(&deg[v], 1.0f);
        atomicAdd(&deg[mx], 1.0f);
    }
}

// ---------------------------------------------------------------------------
// Kernel 6: Laplacian values lap[i] = 1/deg[e0], lap[3F+i] = 1/deg[e1].
// ---------------------------------------------------------------------------
__global__ void lap_kernel(const float* __restrict__ edges_out,
                           const float* __restrict__ deg,
                           const unsigned* __restrict__ Uptr,
                           float* __restrict__ lap, int E3) {
    const int i = blockIdx.x * blockDim.x + threadIdx.x;
    if (i >= E3) return;
    const unsigned U = *Uptr;
    if ((unsigned)i >= U) return;   // padded region already zeroed
    const int e0 = (int)edges_out[2 * (size_t)i + 0];
    const int e1 = (int)edges_out[2 * (size_t)i + 1];
    lap[i]               = 1.0f / deg[e0];
    lap[(size_t)E3 + i]  = 1.0f / deg[e1];
}

// ---------------------------------------------------------------------------
// Kernel 7: normalize vertex normals, inv_areas.
// ---------------------------------------------------------------------------
__global__ void vertex_kernel(float* __restrict__ vn,
                              const float* __restrict__ area_sum,
                              float* __restrict__ inv_areas, int V) {
    const int v = blockIdx.x * blockDim.x + threadIdx.x;
    if (v >= V) return;
    const float x = vn[3 * (size_t)v + 0];
    const float y = vn[3 * (size_t)v + 1];
    const float z = vn[3 * (size_t)v + 2];
    const float nrm = sqrtf(x * x + y * y + z * z);
    const float inv = 1.0f / fmaxf(nrm, 1e-6f);
    vn[3 * (size_t)v + 0] = x * inv;
    vn[3 * (size_t)v + 1] = y * inv;
    vn[3 * (size_t)v + 2] = z * inv;
    const float as = area_sum[v];
    inv_areas[v] = (as > 0.0f) ? (1.0f / as) : as;
}

// ---------------------------------------------------------------------------
// Host launcher (graph-capture safe: kernels + hipMemsetAsync only).
// ---------------------------------------------------------------------------
extern "C" void kernel_launch(void* const* d_in, const int* in_sizes, int n_in,
                              void* d_out, int out_size, void* d_ws, size_t ws_size,
                              hipStream_t stream) {
    const float* verts = (const float*)d_in[0];
    const int*   faces = (const int*)d_in[1];
    const int V  = in_sizes[0] / 3;
    const int F  = in_sizes[1] / 3;
    const int E3 = 3 * F;

    // Output layout (floats, concatenated in reference return order)
    float* out       = (float*)d_out;
    float* vn        = out;
    float* edges     = vn + 3 * (size_t)V;
    float* lap       = edges + 2 * (size_t)E3;
    float* cot       = lap + 2 * (size_t)E3;
    float* inv_areas = cot + 3 * (size_t)F;

    // Workspace carve (256B aligned)
    char* w = (char*)d_ws;
    auto carve = [&](size_t bytes) {
        void* p = (void*)w;
        w += (bytes + 255) & ~(size_t)255;
        return p;
    };
    float*    area_sum = (float*)   carve((size_t)V * 4);
    float*    deg      = (float*)   carve((size_t)V * 4);
    unsigned* counts   = (unsigned*)carve((size_t)V * 4);
    unsigned* offsets  = (unsigned*)carve((size_t)V * 4);
    unsigned* ucount   = (unsigned*)carve((size_t)V * 4);  // doubles as scatter cursor
    unsigned* uoffsets = (unsigned*)carve((size_t)V * 4);
    unsigned* partials = (unsigned*)carve((size_t)8192 * 4);
    unsigned* Utotal   = (unsigned*)carve(256);
    unsigned* smax     = (unsigned*)carve((size_t)E3 * 4);
    (void)ws_size; (void)n_in; (void)out_size;

    // Zero everything that is accumulated or left partially written
    (void)hipMemsetAsync(vn,       0, (size_t)V * 3 * 4, stream);
    (void)hipMemsetAsync(edges,    0, (size_t)E3 * 2 * 4, stream);
    (void)hipMemsetAsync(lap,      0, (size_t)E3 * 2 * 4, stream);
    (void)hipMemsetAsync(area_sum, 0, (size_t)V * 4, stream);
    (void)hipMemsetAsync(deg,      0, (size_t)V * 4, stream);
    (void)hipMemsetAsync(counts,   0, (size_t)V * 4, stream);
    (void)hipMemsetAsync(ucount,   0, (size_t)V * 4, stream);   // cursor for scatter

    const int faceBlocks = (F + FACE_TILE - 1) / FACE_TILE;
    face_kernel<<<faceBlocks, FACE_TILE, 0, stream>>>(verts, faces, vn, cot,
                                                      area_sum, counts, V, F);

    // exclusive scan: counts -> offsets
    const int nTiles = (V + SCAN_TILE - 1) / SCAN_TILE;
    scan_tile_kernel<<<nTiles, SCAN_BLOCK, 0, stream>>>(counts, offsets, partials, V);
    scan_small_kernel<<<1, SCAN_BLOCK, 0, stream>>>(partials, nTiles, nullptr);
    scan_add_kernel<<<(V + 255) / 256, 256, 0, stream>>>(offsets, partials, V);

    scatter_kernel<<<(F + 255) / 256, 256, 0, stream>>>(faces, offsets, ucount, smax, F);

    bucket_kernel<<<(V + 255) / 256, 256, 0, stream>>>(offsets, counts, smax, ucount, V);

    // exclusive scan: ucount -> uoffsets (+ grand total U)
    scan_tile_kernel<<<nTiles, SCAN_BLOCK, 0, stream>>>(ucount, uoffsets, partials, V);
    scan_small_kernel<<<1, SCAN_BLOCK, 0, stream>>>(partials, nTiles, Utotal);
    scan_add_kernel<<<(V + 255) / 256, 256, 0, stream>>>(uoffsets, partials, V);

    emit_kernel<<<(V + 255) / 256, 256, 0, stream>>>(offsets, ucount, uoffsets,
                                                     smax, edges, deg, V);

    lap_kernel<<<(E3 + 255) / 256, 256, 0, stream>>>(edges, deg, Utotal, lap, E3);

    vertex_kernel<<<(V + 255) / 256, 256, 0, stream>>>(vn, area_sum, inv_areas, V);
}